// FastRP_81269371175088
// MI455X (gfx1250) — compile-verified
//
#include <hip/hip_runtime.h>
#include <math.h>

#define DIM 128

typedef __attribute__((ext_vector_type(2))) float v2f;
typedef __attribute__((ext_vector_type(8))) float v8f;

// ---------------------------------------------------------------- zero fill
__global__ void k_zero(float* __restrict__ p, long n) {
    long i = (long)blockIdx.x * blockDim.x + threadIdx.x;
    if (i < n) p[i] = 0.0f;
}

// ------------------------------------------------- degree + per-row counts
__global__ void k_deg_cnt(const int* __restrict__ row, const float* __restrict__ val,
                          float* __restrict__ deg, int* __restrict__ cnt, int e) {
    int i = blockIdx.x * blockDim.x + threadIdx.x;
    if (i < e) {
        int r = row[i];
        atomicAdd(&deg[r], val[i]);
        atomicAdd(&cnt[r], 1);
    }
}

// ---------------------------------- single-block exclusive scan -> row_ptr
__global__ void k_scan(const int* __restrict__ cnt, int* __restrict__ row_ptr, int n) {
    __shared__ int sh[1024];
    __shared__ int running;
    int tid = threadIdx.x;
    if (tid == 0) running = 0;
    __syncthreads();
    for (int base = 0; base < n; base += 1024) {
        int i = base + tid;
        int v = (i < n) ? cnt[i] : 0;
        sh[tid] = v;
        __syncthreads();
        for (int off = 1; off < 1024; off <<= 1) {
            int t = (tid >= off) ? sh[tid - off] : 0;
            __syncthreads();
            sh[tid] += t;
            __syncthreads();
        }
        if (i < n) row_ptr[i] = running + sh[tid] - v;
        __syncthreads();
        if (tid == 0) running += sh[1023];
        __syncthreads();
    }
    if (tid == 0) row_ptr[n] = running;
}

// ------------------- scatter edges into CSR slots, packing (col,val) as int2
__global__ void k_scatter(const int* __restrict__ row, const int* __restrict__ col,
                          const float* __restrict__ val, const int* __restrict__ rptr,
                          int* __restrict__ fill, int2* __restrict__ cpack, int e) {
    int i = blockIdx.x * blockDim.x + threadIdx.x;
    if (i < e) {
        int r = row[i];
        int pos = rptr[r] + atomicAdd(&fill[r], 1);
        int2 p;
        p.x = col[i];
        p.y = __float_as_int(val[i]);
        cpack[pos] = p;
    }
}

// ------------------------------------------------ x0 = R * deg^(-1/2)
__global__ void k_init_x(const float* __restrict__ R, const float* __restrict__ deg,
                         float* __restrict__ x, long nd) {
    long i = (long)blockIdx.x * blockDim.x + threadIdx.x;
    if (i < nd) {
        int r = (int)(i >> 7);
        float d = deg[r];
        if (d == 0.0f) d = 1.0f;
        x[i] = R[i] * (1.0f / sqrtf(d));
    }
}

// ---- fused SpMM + inv_deg scale + row L2-norm + weighted accumulate.
// One wave32 per row; lane l owns floats [4l,4l+4) => 512B/row, coalesced.
// Edge metadata: one coalesced b64 load per lane per 32 edges, wave-broadcast
// via __shfl (ds_bpermute, no memory traffic) instead of uniform VMEM loads.
__global__ void k_spmm(const int* __restrict__ rptr, const int2* __restrict__ cpack,
                       const float* __restrict__ deg,
                       const float* __restrict__ xin, float* __restrict__ xout,
                       float* __restrict__ acc, float w, int n) {
    int wid  = (int)((blockIdx.x * (long)blockDim.x + threadIdx.x) >> 5);
    int lane = threadIdx.x & 31;
    if (wid >= n) return;
    int s = rptr[wid], e = rptr[wid + 1];
    float4 sum = make_float4(0.f, 0.f, 0.f, 0.f);
    for (int base = s; base < e; base += 32) {
        int idx = base + lane;
        int2 pk = make_int2(0, 0);
        if (idx < e) pk = cpack[idx];                 // coalesced 8B per lane
        int m = e - base;
        if (m > 32) m = 32;
        for (int j = 0; j < m; ++j) {
            int   c = __shfl(pk.x, j, 32);            // broadcast edge j
            float v = __int_as_float(__shfl(pk.y, j, 32));
            float4 xv = ((const float4*)(xin + (long)c * DIM))[lane];
            sum.x += v * xv.x; sum.y += v * xv.y;
            sum.z += v * xv.z; sum.w += v * xv.w;
        }
    }
    float d = deg[wid];
    if (d == 0.0f) d = 1.0f;
    float id = 1.0f / d;
    sum.x *= id; sum.y *= id; sum.z *= id; sum.w *= id;
    ((float4*)(xout + (long)wid * DIM))[lane] = sum;   // input to next iter
    // wave32 reduction of squared norm
    float loc = sum.x * sum.x + sum.y * sum.y + sum.z * sum.z + sum.w * sum.w;
    for (int off = 16; off > 0; off >>= 1) loc += __shfl_xor(loc, off, 32);
    float nrm = sqrtf(loc);
    if (nrm < 1e-12f) nrm = 1e-12f;
    float inw = w / nrm;
    float4* pa = (float4*)(acc + (long)wid * DIM) + lane;
    float4 a = *pa;
    a.x += sum.x * inw; a.y += sum.y * inw;
    a.z += sum.z * inw; a.w += sum.w * inw;
    *pa = a;
}

// ---- column sum + sumsq via WMMA: ones(16x4) x B(4x16) -> every D row = colsum.
// Wave w owns columns [16w,16w+16); 4 rows per WMMA pair (B, B^2).
// Steady-state loop is guard-free (block range is uniform); at most one
// guarded tail group. A == ones makes the result K-striping invariant.
__global__ void k_colstats(const float* __restrict__ acc, float* __restrict__ col_sum,
                           float* __restrict__ col_sq, int n, int rows_per_block) {
    int wave  = threadIdx.x >> 5;        // 0..7 -> 16-column chunk
    int lane  = threadIdx.x & 31;
    int nn    = lane & 15;               // column within chunk
    int khalf = (lane >> 4) << 1;        // 0 or 2 (K half, mirrors A layout)
    int colbase = wave * 16;
    int r0 = blockIdx.x * rows_per_block;
    int r1 = r0 + rows_per_block;
    if (r1 > n) r1 = n;
    int rfull = r0 + (((r1 - r0) >> 2) << 2);   // end of full 4-row groups

    const float* bp = acc + colbase + nn;
    v2f a;  a[0] = 1.0f; a[1] = 1.0f;    // all-ones A matrix
    v8f c  = {};
    v8f c2 = {};
    for (int r = r0; r < rfull; r += 4) {
        long o = (long)(r + khalf) * DIM;
        float b0 = bp[o];                // row r+khalf
        float b1 = bp[o + DIM];          // row r+khalf+1
        v2f b;  b[0]  = b0;      b[1]  = b1;
        v2f bq; bq[0] = b0 * b0; bq[1] = b1 * b1;
        c  = __builtin_amdgcn_wmma_f32_16x16x4_f32(false, a, false, b,
                                                   (short)0, c,  false, false);
        c2 = __builtin_amdgcn_wmma_f32_16x16x4_f32(false, a, false, bq,
                                                   (short)0, c2, false, false);
    }
    if (rfull < r1) {                     // block-uniform tail condition
        int ra = rfull + khalf;
        int rb = ra + 1;
        float b0 = (ra < r1) ? bp[(long)ra * DIM] : 0.0f;
        float b1 = (rb < r1) ? bp[(long)rb * DIM] : 0.0f;
        v2f b;  b[0]  = b0;      b[1]  = b1;
        v2f bq; bq[0] = b0 * b0; bq[1] = b1 * b1;
        c  = __builtin_amdgcn_wmma_f32_16x16x4_f32(false, a, false, b,
                                                   (short)0, c,  false, false);
        c2 = __builtin_amdgcn_wmma_f32_16x16x4_f32(false, a, false, bq,
                                                   (short)0, c2, false, false);
    }
    if (lane < 16) {
        atomicAdd(&col_sum[colbase + nn], c[0]);   // D row 0 = column sums
        atomicAdd(&col_sq [colbase + nn], c2[0]);
    }
}

// ---- mean + 1/std (unbiased, std==0 -> 1), stored back into the two arrays
__global__ void k_stats_final(float* __restrict__ col_sum, float* __restrict__ col_sq,
                              float n) {
    int j = threadIdx.x;                 // 128 threads
    float s = col_sum[j], q = col_sq[j];
    float mean = s / n;
    float var  = (q - s * s / n) / (n - 1.0f);
    if (var < 0.0f) var = 0.0f;
    float sd = sqrtf(var);
    if (sd == 0.0f) sd = 1.0f;
    col_sum[j] = mean;
    col_sq[j]  = 1.0f / sd;
}

// ---- out = (out - mean) * inv_std, in place
__global__ void k_finalize(float* __restrict__ out, const float* __restrict__ mean,
                           const float* __restrict__ inv_sd, long nd) {
    long i = (long)blockIdx.x * blockDim.x + threadIdx.x;
    if (i < nd) {
        int j = (int)(i & (DIM - 1));
        out[i] = (out[i] - mean[j]) * inv_sd[j];
    }
}

extern "C" void kernel_launch(void* const* d_in, const int* in_sizes, int n_in,
                              void* d_out, int out_size, void* d_ws, size_t ws_size,
                              hipStream_t stream) {
    const int*   edge_row = (const int*)d_in[0];
    const int*   edge_col = (const int*)d_in[1];
    const float* edge_val = (const float*)d_in[2];
    const float* R        = (const float*)d_in[3];
    const int  E  = in_sizes[0];
    const int  N  = in_sizes[3] / DIM;
    const long ND = (long)N * DIM;
    float* out = (float*)d_out;

    char* ws = (char*)d_ws;
    auto alloc = [&](size_t bytes) -> char* {
        char* p = ws;
        ws += (bytes + 255) & ~(size_t)255;
        return p;
    };
    float* deg   = (float*)alloc((size_t)N * 4);
    int*   cnt   = (int*)  alloc((size_t)N * 4);
    int*   rptr  = (int*)  alloc(((size_t)N + 1) * 4);
    int*   fill  = (int*)  alloc((size_t)N * 4);
    int2*  cpack = (int2*) alloc((size_t)E * 8);
    float* x0    = (float*)alloc((size_t)ND * 4);
    float* x1    = (float*)alloc((size_t)ND * 4);
    float* csum  = (float*)alloc(128 * 4);
    float* csq   = (float*)alloc(128 * 4);
    (void)ws_size; (void)n_in; (void)out_size;

    const int TB = 256;
    k_zero<<<(N + TB - 1) / TB, TB, 0, stream>>>(deg, N);
    k_zero<<<(N + TB - 1) / TB, TB, 0, stream>>>((float*)cnt, N);
    k_zero<<<(N + TB - 1) / TB, TB, 0, stream>>>((float*)fill, N);
    k_zero<<<1, TB, 0, stream>>>(csum, 128);
    k_zero<<<1, TB, 0, stream>>>(csq, 128);
    k_zero<<<(int)((ND + TB - 1) / TB), TB, 0, stream>>>(out, ND);

    k_deg_cnt<<<(E + TB - 1) / TB, TB, 0, stream>>>(edge_row, edge_val, deg, cnt, E);
    k_scan<<<1, 1024, 0, stream>>>(cnt, rptr, N);
    k_scatter<<<(E + TB - 1) / TB, TB, 0, stream>>>(edge_row, edge_col, edge_val,
                                                    rptr, fill, cpack, E);
    k_init_x<<<(int)((ND + TB - 1) / TB), TB, 0, stream>>>(R, deg, x0, ND);

    const float W[4] = {1.0f, 1.0f, 7.81f, 45.28f};
    float* xi = x0;
    float* xo = x1;
    int spmm_blocks = (int)(((long)N * 32 + TB - 1) / TB);   // one wave32 per row
    for (int k = 0; k < 4; ++k) {
        k_spmm<<<spmm_blocks, TB, 0, stream>>>(rptr, cpack, deg,
                                               xi, xo, out, W[k], N);
        float* t = xi; xi = xo; xo = t;
    }

    const int RPB = 512;                                     // rows per stats block
    k_colstats<<<(N + RPB - 1) / RPB, TB, 0, stream>>>(out, csum, csq, N, RPB);
    k_stats_final<<<1, 128, 0, stream>>>(csum, csq, (float)N);
    k_finalize<<<(int)((ND + TB - 1) / TB), TB, 0, stream>>>(out, csum, csq, ND);
}